// Pytorch3dRasterizer_14645838479426
// MI455X (gfx1250) — compile-verified
//
#include <hip/hip_runtime.h>
#include <cstdint>

// Problem constants (from reference): N=8, H=512, W=512, K=1, F=100000, D=16
constexpr uint32_t kN  = 8;
constexpr uint32_t kH  = 512;
constexpr uint32_t kW  = 512;
constexpr uint32_t kD  = 16;
constexpr uint32_t kHW = kH * kW;          // 262144 = 2^18
constexpr uint32_t kP  = kN * kHW;         // 2097152 pixels (K == 1)
constexpr uint32_t kChan = kD + 1;         // 17 output channels

// ---------------------------------------------------------------------------
// Main kernel: one thread per pixel.
//   attrs : [N*F, 3, 16] f32  (153.6 MB table -> keep RT-cached, fits in 192MB L2)
//   p2f   : [P] i32           (streamed once -> NT load)
//   bary  : [P, 3] f32        (streamed once -> NT load)
//   out   : [N, 17, H, W] f32 (streamed once -> NT store, protect L2)
// ---------------------------------------------------------------------------
__global__ __launch_bounds__(256) void raster_interp_kernel(
    const float* __restrict__ attrs,
    const int*   __restrict__ p2f,
    const float* __restrict__ bary,
    float*       __restrict__ out)
{
    const uint32_t tid = blockIdx.x * 256u + threadIdx.x;   // 0 .. kP-1 exact

    // Face index: streamed, non-temporal.
    const int  face  = __builtin_nontemporal_load(p2f + tid);
    const bool valid = (face >= 0);
    const uint32_t idx = valid ? (uint32_t)face : 0u;       // clamp -1 -> 0

    // Gather base: 12 x float4 = 192B per face, 16B aligned.
    const float4* __restrict__ a =
        reinterpret_cast<const float4*>(attrs) + (uint64_t)idx * 12u;

    // Kick the cachelines as soon as the index is known (global_prefetch_b8).
    __builtin_prefetch((const void*)a, 0, 3);
    __builtin_prefetch((const void*)(reinterpret_cast<const char*>(a) + 128), 0, 3);

    // Barycentrics: streamed, non-temporal. Zero them when invalid so the
    // weighted sum is exactly 0 (matches reference's masked-where).
    const uint64_t b3 = (uint64_t)tid * 3u;
    float b0 = __builtin_nontemporal_load(bary + b3 + 0);
    float b1 = __builtin_nontemporal_load(bary + b3 + 1);
    float b2 = __builtin_nontemporal_load(bary + b3 + 2);
    if (!valid) { b0 = 0.0f; b1 = 0.0f; b2 = 0.0f; }

    // 12 x global_load_b128 gather, all in flight before the FMAs.
    float4 v[12];
#pragma unroll
    for (int j = 0; j < 12; ++j) v[j] = a[j];

    // out[d] = b0*v0[d] + b1*v1[d] + b2*v2[d]
    float res[16];
#pragma unroll
    for (int j = 0; j < 4; ++j) {
        res[4 * j + 0] = fmaf(b0, v[j].x, fmaf(b1, v[4 + j].x, b2 * v[8 + j].x));
        res[4 * j + 1] = fmaf(b0, v[j].y, fmaf(b1, v[4 + j].y, b2 * v[8 + j].y));
        res[4 * j + 2] = fmaf(b0, v[j].z, fmaf(b1, v[4 + j].z, b2 * v[8 + j].z));
        res[4 * j + 3] = fmaf(b0, v[j].w, fmaf(b1, v[4 + j].w, b2 * v[8 + j].w));
    }

    // Output layout [N, 17, H, W]: consecutive lanes -> consecutive w, so each
    // per-channel NT store is a fully coalesced 128B/wave transaction.
    const uint32_t n  = tid >> 18;            // tid / kHW
    const uint32_t hw = tid & (kHW - 1u);
    float* __restrict__ o = out + (uint64_t)n * kChan * kHW + hw;

#pragma unroll
    for (int d = 0; d < 16; ++d)
        __builtin_nontemporal_store(res[d], o + (uint64_t)d * kHW);
    __builtin_nontemporal_store(valid ? 1.0f : 0.0f, o + (uint64_t)16 * kHW);
}

// ---------------------------------------------------------------------------
// Probe kernels (never launched): verify gfx1250 async/TDM asm spellings on
// this toolchain so a TDM-based output path can be wired in a later round.
// ---------------------------------------------------------------------------
__global__ void __launch_bounds__(32) cdna5_probe_async_lds(
    const float4* __restrict__ src, float4* __restrict__ dst)
{
    __shared__ float4 buf[32];                 // only shared var -> LDS offset 0
    const unsigned lane   = threadIdx.x & 31u;
    const unsigned ldsOff = lane * 16u;        // byte offset into wave LDS
    const float4*  g      = src + lane;

    asm volatile("global_load_async_to_lds_b128 %0, %1, off"
                 :: "v"(ldsOff), "v"(g)
                 : "memory");
    asm volatile("s_wait_asynccnt 0" ::: "memory");

    dst[lane] = buf[lane];
}

typedef __attribute__((ext_vector_type(4))) unsigned tdm_g0_t;
typedef __attribute__((ext_vector_type(8))) unsigned tdm_g1_t;

__global__ void cdna5_probe_tdm()
{
    tdm_g0_t g0 = {};
    tdm_g1_t g1 = {};
    asm volatile("tensor_load_to_lds %0, %1"
                 :: "s"(g0), "s"(g1)
                 : "memory");
    __builtin_amdgcn_s_wait_tensorcnt(0);
}

// ---------------------------------------------------------------------------
// Launch
// ---------------------------------------------------------------------------
extern "C" void kernel_launch(void* const* d_in, const int* in_sizes, int n_in,
                              void* d_out, int out_size, void* d_ws, size_t ws_size,
                              hipStream_t stream) {
    (void)in_sizes; (void)n_in; (void)out_size; (void)d_ws; (void)ws_size;

    const float* attrs = (const float*)d_in[0];   // attributes [N,F,3,D]
    const int*   p2f   = (const int*)  d_in[1];   // pix_to_face [N,H,W,K]
    const float* bary  = (const float*)d_in[2];   // bary_coords [N,H,W,K,3]
    float*       out   = (float*)d_out;           // [N, D+1, H, W]

    dim3 grid(kP / 256u);
    dim3 block(256u);
    hipLaunchKernelGGL(raster_interp_kernel, grid, block, 0, stream,
                       attrs, p2f, bary, out);
}